// GNNTC_45938970198063
// MI455X (gfx1250) — compile-verified
//
#include <hip/hip_runtime.h>

// Problem constants (from reference)
#define BB 4
#define LL 64
#define NN 23
#define HH 256
#define NF 256
#define MA 92        // B*N rows
#define MAP 96       // padded to multiple of 32
#define MS 5888      // B*L*N rows (multiple of 32)
#define MP 256       // B*L rows  (multiple of 32)

typedef float v2f __attribute__((ext_vector_type(2)));
typedef float v8f __attribute__((ext_vector_type(8)));

// ---------------------------------------------------------------------------
// Small-K dense layer (K=3): out = relu(A @ W + b), zero-filling pad rows so
// downstream WMMA GEMMs need no M bounds checks.
// ---------------------------------------------------------------------------
__global__ void smallk_gemm_relu(const float* __restrict__ A,
                                 const float* __restrict__ W,
                                 const float* __restrict__ bias,
                                 float* __restrict__ C,
                                 int M, int Mpad, int K, int N) {
  int idx = blockIdx.x * blockDim.x + threadIdx.x;
  if (idx >= Mpad * N) return;
  int m = idx / N, n = idx - m * N;
  if (m >= M) { C[idx] = 0.f; return; }
  float s = bias[n];
  for (int k = 0; k < K; ++k) s = fmaf(A[(size_t)m * K + k], W[(size_t)k * N + n], s);
  C[idx] = fmaxf(s, 0.f);
}

// ---------------------------------------------------------------------------
// 256x256 transpose (LDS-tiled): out[n*256+k] = in[k*256+n]
// Lets the GEMM load B fragments as contiguous float2 per lane.
// ---------------------------------------------------------------------------
__global__ void transpose256(const float* __restrict__ in, float* __restrict__ out) {
  __shared__ float tile[16][17];
  const int x = blockIdx.x * 16 + threadIdx.x;   // col of in
  const int y = blockIdx.y * 16 + threadIdx.y;   // row of in
  tile[threadIdx.y][threadIdx.x] = in[y * 256 + x];
  __syncthreads();
  const int ox = blockIdx.y * 16 + threadIdx.x;  // col of out (= in row)
  const int oy = blockIdx.x * 16 + threadIdx.y;  // row of out (= in col)
  out[oy * 256 + ox] = tile[threadIdx.x][threadIdx.y];
}

// ---------------------------------------------------------------------------
// Fused GEMM via V_WMMA_F32_16X16X4_F32: C = act(A[MxK] @ W + bias), with W
// supplied transposed (Wt is N x K).  One wave computes a 32x32 C tile
// (2x2 register blocking); K marched in steps of 4 with a software-pipelined
// (double-buffered) fragment prefetch so loads for chunk k+4 are in flight
// while the WMMAs for chunk k execute.  M, N multiples of 32; no bounds
// checks, EXEC stays all-ones (WMMA requirement).
// Fragment layouts per CDNA5 ISA §7.12.2:
//   A (16x4):  lane l<16 -> row l, {K0,K1}; lane l>=16 -> row l-16, {K2,K3}
//   B (4x16):  symmetric with N = lane%16 (via the transposed weight)
//   C (16x16): VGPR v -> row v (lanes 0-15) / row v+8 (lanes 16-31)
// ---------------------------------------------------------------------------
__global__ void __launch_bounds__(32)
gemm_wmma_f32(const float* __restrict__ A,
              const float* __restrict__ Wt,   // N x K (transposed weight)
              const float* __restrict__ bias,
              float* __restrict__ C,
              int K, int N, int relu) {
  const int lane = threadIdx.x;     // 32 threads = 1 wave
  const int l16  = lane & 15;
  const int half = lane >> 4;
  const int mt   = blockIdx.x * 32;
  const int nt   = blockIdx.y * 32;

  const float* A0 = A  + (size_t)(mt + l16)      * K + half * 2;
  const float* A1 = A  + (size_t)(mt + 16 + l16) * K + half * 2;
  const float* B0 = Wt + (size_t)(nt + l16)      * K + half * 2;
  const float* B1 = Wt + (size_t)(nt + 16 + l16) * K + half * 2;

  v8f acc00 = {0.f,0.f,0.f,0.f,0.f,0.f,0.f,0.f};
  v8f acc01 = acc00, acc10 = acc00, acc11 = acc00;

  // prologue: fragments for chunk 0
  v2f a0c = *(const v2f*)(A0);
  v2f a1c = *(const v2f*)(A1);
  v2f b0c = *(const v2f*)(B0);
  v2f b1c = *(const v2f*)(B1);

#pragma unroll 4
  for (int k0 = 4; k0 < K; k0 += 4) {
    // issue next chunk's loads before consuming the current fragments
    const v2f a0n = *(const v2f*)(A0 + k0);
    const v2f a1n = *(const v2f*)(A1 + k0);
    const v2f b0n = *(const v2f*)(B0 + k0);
    const v2f b1n = *(const v2f*)(B1 + k0);
    acc00 = __builtin_amdgcn_wmma_f32_16x16x4_f32(false, a0c, false, b0c, (short)0, acc00, false, false);
    acc01 = __builtin_amdgcn_wmma_f32_16x16x4_f32(false, a0c, false, b1c, (short)0, acc01, false, false);
    acc10 = __builtin_amdgcn_wmma_f32_16x16x4_f32(false, a1c, false, b0c, (short)0, acc10, false, false);
    acc11 = __builtin_amdgcn_wmma_f32_16x16x4_f32(false, a1c, false, b1c, (short)0, acc11, false, false);
    a0c = a0n; a1c = a1n; b0c = b0n; b1c = b1n;
  }
  // epilogue: last chunk
  acc00 = __builtin_amdgcn_wmma_f32_16x16x4_f32(false, a0c, false, b0c, (short)0, acc00, false, false);
  acc01 = __builtin_amdgcn_wmma_f32_16x16x4_f32(false, a0c, false, b1c, (short)0, acc01, false, false);
  acc10 = __builtin_amdgcn_wmma_f32_16x16x4_f32(false, a1c, false, b0c, (short)0, acc10, false, false);
  acc11 = __builtin_amdgcn_wmma_f32_16x16x4_f32(false, a1c, false, b1c, (short)0, acc11, false, false);

  const float bv0 = bias ? bias[nt + l16]      : 0.f;
  const float bv1 = bias ? bias[nt + 16 + l16] : 0.f;
  const int   r0  = mt + half * 8;
  const int   r1  = mt + 16 + half * 8;
  const int   c0  = nt + l16;
  const int   c1  = nt + 16 + l16;
#pragma unroll
  for (int v = 0; v < 8; ++v) {
    float e00 = acc00[v] + bv0, e01 = acc01[v] + bv1;
    float e10 = acc10[v] + bv0, e11 = acc11[v] + bv1;
    if (relu) {
      e00 = fmaxf(e00, 0.f); e01 = fmaxf(e01, 0.f);
      e10 = fmaxf(e10, 0.f); e11 = fmaxf(e11, 0.f);
    }
    C[(size_t)(r0 + v) * N + c0] = e00;
    C[(size_t)(r0 + v) * N + c1] = e01;
    C[(size_t)(r1 + v) * N + c0] = e10;
    C[(size_t)(r1 + v) * N + c1] = e11;
  }
}

// ---------------------------------------------------------------------------
// Gather state_enc rows at i = playerid[b]:  senc_p[(b*L+l), c]
// ---------------------------------------------------------------------------
__global__ void gather_player(const float* __restrict__ state_enc,
                              const int* __restrict__ pid,
                              float* __restrict__ out) {
  const int bl = blockIdx.x;            // 0..B*L-1
  const int c  = threadIdx.x;           // 0..H-1
  const int b  = bl / LL;
  const int p  = pid[b];
  out[(size_t)bl * HH + c] = state_enc[((size_t)bl * NN + p) * HH + c];
}

// ---------------------------------------------------------------------------
// h1_p[b*N+j, c] = relu(pa[b, p_b, c] + pb[b, j, c] + re_b1[c]); pad rows = 0
// ---------------------------------------------------------------------------
__global__ void h1_player(const float* __restrict__ pa,
                          const float* __restrict__ pb,
                          const float* __restrict__ re_b1,
                          const int* __restrict__ pid,
                          float* __restrict__ out) {
  const int bj = blockIdx.x;            // 0..MAP-1
  const int c  = threadIdx.x;
  if (bj >= MA) { out[(size_t)bj * HH + c] = 0.f; return; }
  const int b  = bj / NN;
  const int p  = pid[b];
  const float v = pa[((size_t)b * NN + p) * HH + c] + pb[(size_t)bj * HH + c] + re_b1[c];
  out[(size_t)bj * HH + c] = fmaxf(v, 0.f);
}

// ---------------------------------------------------------------------------
// x[b, c, l] = sum_j relu(ps_p[b*L+l, c] + pr_r[(b*L+l)*N+j, c]
//                         + prel_p[b*N+j, c] + pr_b[c])   (stored transposed)
// ---------------------------------------------------------------------------
__global__ void combine_effect(const float* __restrict__ ps_p,
                               const float* __restrict__ pr_r,
                               const float* __restrict__ prel_p,
                               const float* __restrict__ pr_b,
                               float* __restrict__ xT) {
  const int bl = blockIdx.x;            // 0..B*L-1
  const int c  = threadIdx.x;
  const int b  = bl / LL;
  const int l  = bl - b * LL;
  const float base = ps_p[(size_t)bl * HH + c] + pr_b[c];
  float s = 0.f;
  for (int j = 0; j < NN; ++j) {
    const float v = base + pr_r[((size_t)bl * NN + j) * HH + c]
                         + prel_p[((size_t)b * NN + j) * HH + c];
    s += fmaxf(v, 0.f);
  }
  xT[((size_t)b * HH + c) * LL + l] = s;
}

// ---------------------------------------------------------------------------
// 1-D conv, NCH layout, one wave per output element, warp-reduce over Cin.
// ---------------------------------------------------------------------------
__global__ void conv1d_wave(const float* __restrict__ x,
                            const float* __restrict__ w,
                            const float* __restrict__ bias,
                            float* __restrict__ y,
                            int Cin, int Lin, int Cout, int Lout,
                            int Kw, int stride, int pad, int relu) {
  const int gtid = blockIdx.x * blockDim.x + threadIdx.x;
  const int widx = gtid >> 5;
  const int lane = gtid & 31;
  const int total = BB * Cout * Lout;
  if (widx >= total) return;
  const int t  = widx % Lout;
  const int oc = (widx / Lout) % Cout;
  const int b  = widx / (Lout * Cout);
  const int base = t * stride - pad;
  float s = 0.f;
  for (int ic = lane; ic < Cin; ic += 32) {
    const float* xr = x + ((size_t)b * Cin + ic) * Lin;
    const float* wr = w + ((size_t)oc * Cin + ic) * Kw;
    for (int k = 0; k < Kw; ++k) {
      const int p = base + k;
      if (p >= 0 && p < Lin) s = fmaf(xr[p], wr[k], s);
    }
  }
#pragma unroll
  for (int off = 16; off > 0; off >>= 1) s += __shfl_down(s, off, 32);
  if (lane == 0) {
    s += bias[oc];
    if (relu) s = fmaxf(s, 0.f);
    y[widx] = s;
  }
}

// ---------------------------------------------------------------------------
// Final 1x1 conv, evaluated only at center position 3 (all the output reads).
// ---------------------------------------------------------------------------
__global__ void conv4_at3(const float* __restrict__ y3,
                          const float* __restrict__ w,
                          const float* __restrict__ bias,
                          float* __restrict__ out) {
  const int idx = blockIdx.x * blockDim.x + threadIdx.x;  // 0..B*NF-1
  if (idx >= BB * NF) return;
  const int oc = idx % NF;
  const int b  = idx / NF;
  float s = bias[oc];
#pragma unroll
  for (int ic = 0; ic < 32; ++ic)
    s = fmaf(y3[((size_t)b * 32 + ic) * 6 + 3], w[oc * 32 + ic], s);
  out[idx] = s;
}

// ---------------------------------------------------------------------------
extern "C" void kernel_launch(void* const* d_in, const int* in_sizes, int n_in,
                              void* d_out, int out_size, void* d_ws, size_t ws_size,
                              hipStream_t stream) {
  const float* states = (const float*)d_in[0];
  const float* types  = (const float*)d_in[1];
  const int*   pid    = (const int*)  d_in[2];
  const float* ae_W1  = (const float*)d_in[3];
  const float* ae_b1  = (const float*)d_in[4];
  const float* ae_W2  = (const float*)d_in[5];
  const float* ae_b2  = (const float*)d_in[6];
  const float* se_W1  = (const float*)d_in[7];
  const float* se_b1  = (const float*)d_in[8];
  const float* se_W2  = (const float*)d_in[9];
  const float* se_b2  = (const float*)d_in[10];
  const float* re_W1  = (const float*)d_in[11];  // (512,256)
  const float* re_b1  = (const float*)d_in[12];
  const float* re_W2  = (const float*)d_in[13];
  const float* re_b2  = (const float*)d_in[14];
  const float* pr_W   = (const float*)d_in[15];  // (768,256)
  const float* pr_b   = (const float*)d_in[16];
  const float* c1_w   = (const float*)d_in[17];
  const float* c1_b   = (const float*)d_in[18];
  const float* c2_w   = (const float*)d_in[19];
  const float* c2_b   = (const float*)d_in[20];
  const float* c3_w   = (const float*)d_in[21];
  const float* c3_b   = (const float*)d_in[22];
  const float* c4_w   = (const float*)d_in[23];
  const float* c4_b   = (const float*)d_in[24];

  // Workspace carve-up (floats)
  float* ws = (float*)d_ws;
  float* h_a    = ws;                          // MAP*256
  float* aenc   = h_a    + (size_t)MAP * HH;
  float* pb     = aenc   + (size_t)MAP * HH;
  float* pa     = pb     + (size_t)MAP * HH;
  float* h1_p   = pa     + (size_t)MAP * HH;
  float* rel_p  = h1_p   + (size_t)MAP * HH;
  float* prel_p = rel_p  + (size_t)MAP * HH;
  float* h_s    = prel_p + (size_t)MAP * HH;   // MS*256
  float* senc   = h_s    + (size_t)MS * HH;
  float* pr_r   = senc   + (size_t)MS * HH;
  float* senc_p = pr_r   + (size_t)MS * HH;    // MP*256
  float* ps_p   = senc_p + (size_t)MP * HH;
  float* xT     = ps_p   + (size_t)MP * HH;    // B*H*L
  float* y1     = xT     + (size_t)BB * HH * LL;
  float* y2     = y1     + BB * 32 * 22;
  float* y3     = y2     + BB * 32 * 11;
  float* wt     = y3     + BB * 32 * 6;        // 8 transposed 256x256 weights
  float* wt_aeW2  = wt + 0 * HH * HH;
  float* wt_reW1a = wt + 1 * HH * HH;
  float* wt_reW1b = wt + 2 * HH * HH;
  float* wt_seW2  = wt + 3 * HH * HH;
  float* wt_prWb  = wt + 4 * HH * HH;   // pr_W[H:2H]
  float* wt_prWa  = wt + 5 * HH * HH;   // pr_W[0:H]
  float* wt_reW2  = wt + 6 * HH * HH;
  float* wt_prWc  = wt + 7 * HH * HH;   // pr_W[2H:3H]

  const dim3 w32(32);
  const dim3 t16(16, 16);
  const dim3 tg(16, 16);
  const dim3 gA(MAP / 32, HH / 32);   // (3,8)
  const dim3 gS(MS  / 32, HH / 32);   // (184,8)
  const dim3 gP(MP  / 32, HH / 32);   // (8,8)

  // --- weight transposes (Wt = W^T so B fragments load as contiguous b64) ---
  transpose256<<<tg, t16, 0, stream>>>(ae_W2,              wt_aeW2);
  transpose256<<<tg, t16, 0, stream>>>(re_W1,              wt_reW1a);
  transpose256<<<tg, t16, 0, stream>>>(re_W1 + HH * HH,    wt_reW1b);
  transpose256<<<tg, t16, 0, stream>>>(se_W2,              wt_seW2);
  transpose256<<<tg, t16, 0, stream>>>(pr_W + HH * HH,     wt_prWb);
  transpose256<<<tg, t16, 0, stream>>>(pr_W,               wt_prWa);
  transpose256<<<tg, t16, 0, stream>>>(re_W2,              wt_reW2);
  transpose256<<<tg, t16, 0, stream>>>(pr_W + 2 * HH * HH, wt_prWc);

  // --- encoders, K=3 input layers (VALU), pad rows zeroed ---
  smallk_gemm_relu<<<(MAP * HH + 255) / 256, 256, 0, stream>>>(types,  ae_W1, ae_b1, h_a, MA, MAP, 3, HH);
  smallk_gemm_relu<<<(MS  * HH + 255) / 256, 256, 0, stream>>>(states, se_W1, se_b1, h_s, MS, MS,  3, HH);

  // --- K=256 layers (WMMA f32 16x16x4, 32x32 tile per wave) ---
  gemm_wmma_f32<<<gA, w32, 0, stream>>>(h_a,  wt_aeW2,  ae_b2,   aenc, HH, HH, 1);
  gemm_wmma_f32<<<gA, w32, 0, stream>>>(aenc, wt_reW1a, nullptr, pb,   HH, HH, 0);
  gemm_wmma_f32<<<gA, w32, 0, stream>>>(aenc, wt_reW1b, nullptr, pa,   HH, HH, 0);
  gemm_wmma_f32<<<gS, w32, 0, stream>>>(h_s,  wt_seW2,  se_b2,   senc, HH, HH, 1);
  gemm_wmma_f32<<<gS, w32, 0, stream>>>(senc, wt_prWb,  nullptr, pr_r, HH, HH, 0);

  // --- player gather + ps (only rows the output reads) ---
  gather_player<<<MP, HH, 0, stream>>>(senc, pid, senc_p);
  gemm_wmma_f32<<<gP, w32, 0, stream>>>(senc_p, wt_prWa, nullptr, ps_p, HH, HH, 0);

  // --- relation path restricted to i = playerid[b] ---
  h1_player<<<MAP, HH, 0, stream>>>(pa, pb, re_b1, pid, h1_p);
  gemm_wmma_f32<<<gA, w32, 0, stream>>>(h1_p,  wt_reW2, re_b2,   rel_p,  HH, HH, 1);
  gemm_wmma_f32<<<gA, w32, 0, stream>>>(rel_p, wt_prWc, nullptr, prel_p, HH, HH, 0);

  // --- fused effect/relu/sum + transpose to NCH ---
  combine_effect<<<MP, HH, 0, stream>>>(ps_p, pr_r, prel_p, pr_b, xT);

  // --- conv stack (wave-per-output) ---
  {
    int tot = BB * 32 * 22;   // conv1: 256->32, k9 s3 p4
    conv1d_wave<<<(tot * 32 + 255) / 256, 256, 0, stream>>>(xT, c1_w, c1_b, y1, HH, LL, 32, 22, 9, 3, 4, 1);
    tot = BB * 32 * 11;       // conv2: 32->32, k7 s2 p3
    conv1d_wave<<<(tot * 32 + 255) / 256, 256, 0, stream>>>(y1, c2_w, c2_b, y2, 32, 22, 32, 11, 7, 2, 3, 1);
    tot = BB * 32 * 6;        // conv3: 32->32, k5 s2 p2
    conv1d_wave<<<(tot * 32 + 255) / 256, 256, 0, stream>>>(y2, c3_w, c3_b, y3, 32, 11, 32, 6, 5, 2, 2, 1);
  }

  // --- final 1x1 conv at center position only ---
  conv4_at3<<<(BB * NF + 255) / 256, 256, 0, stream>>>(y3, c4_w, c4_b, (float*)d_out);
}